// TSALoss_79852031967238
// MI455X (gfx1250) — compile-verified
//
#include <hip/hip_runtime.h>
#include <hip/hip_bf16.h>

#define NPTS 4096
#define DIM  128
#define KNB  25
#define GP   32          // zero-padded Gram size
#define XCS  132         // Xc LDS row stride (floats), bank-conflict padding
#define GS   33          // Gram LDS row stride
#define PITERS 100
#define BIGF 3.0e38f

typedef __attribute__((ext_vector_type(2))) float v2f;
typedef __attribute__((ext_vector_type(8))) float v8f;

// ---------------- Kernel 1: fused distance row + top-25 selection ----------------
// One 128-thread workgroup per row i. d2 row lives in LDS (16KB), then 25
// sequential argmin extractions (tie-break: smallest index, matching stable top_k).
__global__ void __launch_bounds__(128)
tsa_knn_kernel(const float* __restrict__ raw, int* __restrict__ nbrs)
{
    __shared__ float d2[NPTS];
    __shared__ __align__(16) float ri[DIM];
    __shared__ float redv[128];
    __shared__ int   redi[128];

    const int i = blockIdx.x;
    const int t = threadIdx.x;

    if (t < DIM) ri[t] = raw[i * DIM + t];
    __syncthreads();

    // thread t owns contiguous j-chunk [t*32, t*32+32)
    for (int jj = 0; jj < 32; ++jj) {
        const int j = t * 32 + jj;
        const float4* rj = (const float4*)(raw + (size_t)j * DIM);
        const float4* ra = (const float4*)ri;
        float acc = 0.0f;
        #pragma unroll 8
        for (int k = 0; k < DIM / 4; ++k) {
            float4 a = ra[k];
            float4 b = rj[k];
            float dx = a.x - b.x, dy = a.y - b.y;
            float dz = a.z - b.z, dw = a.w - b.w;
            acc += dx * dx + dy * dy + dz * dz + dw * dw;
        }
        d2[j] = (j == i) ? BIGF : acc;   // exclude self
    }
    __syncthreads();

    for (int sel = 0; sel < KNB; ++sel) {
        float mv = BIGF;
        int   mi = NPTS;
        for (int jj = 0; jj < 32; ++jj) {        // increasing j: strict < keeps first
            const int j = t * 32 + jj;
            const float v = d2[j];
            if (v < mv) { mv = v; mi = j; }
        }
        redv[t] = mv; redi[t] = mi;
        __syncthreads();
        for (int s = 64; s > 0; s >>= 1) {
            if (t < s) {
                const float ov = redv[t + s];
                const int   oi = redi[t + s];
                if (ov < redv[t] || (ov == redv[t] && oi < redi[t])) {
                    redv[t] = ov; redi[t] = oi;
                }
            }
            __syncthreads();
        }
        if (t == 0) {
            nbrs[i * KNB + sel] = redi[0];
            d2[redi[0]] = BIGF;                  // extract
        }
        __syncthreads();
    }
}

// ---------------- Kernel 2: per-point top eigenvector via Gram + WMMA ----------------
// One wave32 per point. For X in {latent, raw}: gather 25 neighbor rows, center,
// G = Xc*Xc^T (32x32 zero-padded) via V_WMMA_F32_16X16X4_F32, power-iterate on G,
// back-project u = Xc^T w, normalize. loss_i = 2 - 2*(u.v)^2.
__global__ void __launch_bounds__(32)
tsa_eig_kernel(const float* __restrict__ latent, const float* __restrict__ raw,
               const int* __restrict__ nbrs, float* __restrict__ losses)
{
    __shared__ float Xc[GP * XCS];   // 32 x 128 (stride 132), rows >= 25 are zero
    __shared__ float G[GP * GS];     // 32 x 32 (stride 33)
    __shared__ float wsh[GP];
    __shared__ float uvec[DIM];
    __shared__ float vvec[DIM];

    const int i    = blockIdx.x;
    const int lane = threadIdx.x;        // 0..31, wave32
    const int half = lane >> 4;          // 0 or 1
    const int l16  = lane & 15;

    for (int pass = 0; pass < 2; ++pass) {
        const float* X   = pass ? raw : latent;
        float*       dst = pass ? vvec : uvec;

        // ---- gather neighbor rows (rows 25..31 zeroed) ----
        for (int m = 0; m < GP; ++m) {
            if (m < KNB) {
                const int r = nbrs[i * KNB + m];     // uniform -> scalar load
                for (int c = lane; c < DIM; c += 32)
                    Xc[m * XCS + c] = X[(size_t)r * DIM + c];
            } else {
                for (int c = lane; c < DIM; c += 32)
                    Xc[m * XCS + c] = 0.0f;
            }
        }
        __syncthreads();

        // ---- center columns over the 25 neighbors ----
        for (int c = lane; c < DIM; c += 32) {
            float s = 0.0f;
            for (int m = 0; m < KNB; ++m) s += Xc[m * XCS + c];
            s *= (1.0f / KNB);
            for (int m = 0; m < KNB; ++m) Xc[m * XCS + c] -= s;
        }
        __syncthreads();

        // ---- G = Xc * Xc^T via f32 WMMA (four 16x16 tiles, K=128) ----
        for (int tr = 0; tr < 2; ++tr) {
            for (int tc = 0; tc < 2; ++tc) {
                v8f acc = {};
                for (int kk = 0; kk < DIM; kk += 4) {
                    const int k0 = kk + half * 2;
                    v2f a, b;
                    // A fragment: 16x4 tile, lane(l16) = row, halves hold K pairs
                    a.x = Xc[(tr * 16 + l16) * XCS + k0];
                    a.y = Xc[(tr * 16 + l16) * XCS + k0 + 1];
                    // B = Xc^T  => mirror load from row block tc
                    b.x = Xc[(tc * 16 + l16) * XCS + k0];
                    b.y = Xc[(tc * 16 + l16) * XCS + k0 + 1];
                    acc = __builtin_amdgcn_wmma_f32_16x16x4_f32(
                        false, a, false, b, (short)0, acc, false, false);
                }
                // C/D layout: VGPR j -> M = j + half*8, N = l16
                #pragma unroll
                for (int j = 0; j < 8; ++j)
                    G[(tr * 16 + j + half * 8) * GS + tc * 16 + l16] = acc[j];
            }
        }
        __syncthreads();

        // ---- power iteration on G (lane r owns w[r]) ----
        float w = (lane < KNB) ? 1.0f : 0.0f;
        for (int it = 0; it < PITERS; ++it) {
            wsh[lane] = w;
            __syncthreads();
            float nw = 0.0f;
            #pragma unroll
            for (int c = 0; c < GP; ++c) nw += G[lane * GS + c] * wsh[c];
            float ss = nw * nw;
            #pragma unroll
            for (int off = 16; off > 0; off >>= 1) ss += __shfl_xor(ss, off, 32);
            w = nw * rsqrtf(ss + 1e-30f);
            __syncthreads();
        }
        wsh[lane] = w;
        __syncthreads();

        // ---- back-project: u = Xc^T w, normalize, stash ----
        float comp[4];
        float ssl = 0.0f;
        #pragma unroll
        for (int ci = 0; ci < 4; ++ci) {
            const int c = lane + ci * 32;
            float s = 0.0f;
            for (int m = 0; m < KNB; ++m) s += Xc[m * XCS + c] * wsh[m];
            comp[ci] = s;
            ssl += s * s;
        }
        #pragma unroll
        for (int off = 16; off > 0; off >>= 1) ssl += __shfl_xor(ssl, off, 32);
        const float inv = rsqrtf(ssl + 1e-30f);
        #pragma unroll
        for (int ci = 0; ci < 4; ++ci) dst[lane + ci * 32] = comp[ci] * inv;
        __syncthreads();
    }

    // ---- loss_i = ||uu^T - vv^T||_F^2 = 2 - 2 (u.v)^2 ----
    float d = 0.0f;
    #pragma unroll
    for (int ci = 0; ci < 4; ++ci)
        d += uvec[lane + ci * 32] * vvec[lane + ci * 32];
    #pragma unroll
    for (int off = 16; off > 0; off >>= 1) d += __shfl_xor(d, off, 32);
    if (lane == 0) losses[i] = 2.0f - 2.0f * d * d;
}

// ---------------- Kernel 3: deterministic reduction ----------------
__global__ void __launch_bounds__(256)
tsa_reduce_kernel(const float* __restrict__ losses, float* __restrict__ out)
{
    __shared__ float s[256];
    const int t = threadIdx.x;
    float a = 0.0f;
    for (int idx = t; idx < NPTS; idx += 256) a += losses[idx];
    s[t] = a;
    __syncthreads();
    for (int st = 128; st > 0; st >>= 1) {
        if (t < st) s[t] += s[t + st];
        __syncthreads();
    }
    if (t == 0) out[0] = 0.1f * s[0] * (1.0f / (float)NPTS);
}

extern "C" void kernel_launch(void* const* d_in, const int* in_sizes, int n_in,
                              void* d_out, int out_size, void* d_ws, size_t ws_size,
                              hipStream_t stream)
{
    const float* latent = (const float*)d_in[0];
    const float* raw    = (const float*)d_in[1];
    float* out = (float*)d_out;

    int*   nbrs   = (int*)d_ws;                                        // 4096*25*4 B
    float* losses = (float*)((char*)d_ws + (size_t)NPTS * KNB * sizeof(int));

    tsa_knn_kernel<<<NPTS, 128, 0, stream>>>(raw, nbrs);
    tsa_eig_kernel<<<NPTS, 32, 0, stream>>>(latent, raw, nbrs, losses);
    tsa_reduce_kernel<<<1, 256, 0, stream>>>(losses, out);
}